// FFTConv_12025908428814
// MI455X (gfx1250) — compile-verified
//
#include <hip/hip_runtime.h>
#include <hip/hip_bf16.h>

// FFT conv as three WMMA-f32 stages on gfx1250.
// ws layout (floats):
//   [0, 8704)              : DFT matrix, padded Fr[64][68] then Fi[64][68]
//   XR: 3136*2048 floats   : forward spectrum real, layout [f][b*64+i]
//   XI: 3136*2048
//   YR: 4096*3136 floats   : product spectrum real, layout [(b*128+o)][f]
//   YI: 4096*3136
// total ~154.2 MB of workspace.

typedef __attribute__((ext_vector_type(2))) float v2f;
typedef __attribute__((ext_vector_type(8))) float v8f;

#define NP   64          // padded matrix dim
#define LDP  68          // LDS row stride (floats), avoids bank conflicts
#define NPAD (NP * LDP)  // 4352 floats per padded 64x64 matrix
#define NS   56          // true spatial dim
#define NF   3136        // 56*56 frequencies
#define NIMG 2048        // B*Cin
#define NOUT 4096        // B*Cout

static __device__ __forceinline__ v8f zero8() {
  v8f z = {0.f, 0.f, 0.f, 0.f, 0.f, 0.f, 0.f, 0.f};
  return z;
}

// D += A(16x64) * B(64x16) as 16 chained V_WMMA_F32_16X16X4_F32.
// A: row-major [m][k], lda stride; B: row-major [k][n], ldb stride.
// Wave32 layouts per CDNA5 ISA 7.12.2:
//   A 16x4 : lane(0-15)=M, VGPR j holds K = j + 2*(lane>>4)
//   B 4x16 : lane(0-15)=N, VGPR j holds K = j + 2*(lane>>4)
static __device__ __forceinline__ v8f wmma_chain64(const float* __restrict__ A, int lda,
                                                   const float* __restrict__ Bm, int ldb,
                                                   v8f acc) {
  const int l = threadIdx.x & 31;
  const int mn = l & 15;
  const int kh = (l >> 4) << 1;
#pragma unroll
  for (int k = 0; k < 64; k += 4) {
    v2f a, b;
    a[0] = A[mn * lda + k + kh];
    a[1] = A[mn * lda + k + kh + 1];
    b[0] = Bm[(k + kh) * ldb + mn];
    b[1] = Bm[(k + kh + 1) * ldb + mn];
    acc = __builtin_amdgcn_wmma_f32_16x16x4_f32(false, a, false, b, (short)0, acc,
                                                false, false);
  }
  return acc;
}

// Store 16x16 f32 D tile into LDS. C/D layout: VGPR v -> row v + 8*(lane>>4), col lane&15.
static __device__ __forceinline__ void stD_lds(float* base, int ld, v8f d) {
  const int l = threadIdx.x & 31;
  const int n = l & 15;
  const int mh = (l >> 4) << 3;
#pragma unroll
  for (int v = 0; v < 8; ++v) base[(mh + v) * ld + n] = d[v];
}

// -------- stage 0: DFT matrix (padded 64x68, zero-filled outside 56x56) --------
__global__ void build_dft_kernel(float* __restrict__ dft) {
  int idx = blockIdx.x * 256 + threadIdx.x;
  if (idx >= NPAD) return;
  int r = idx / LDP, c = idx - r * LDP;
  float vr = 0.f, vi = 0.f;
  if (r < NS && c < NS) {
    float ang = -2.0f * 3.14159265358979323846f * (float)((r * c) % NS) / (float)NS;
    vr = __builtin_cosf(ang);
    vi = __builtin_sinf(ang);
  }
  dft[idx] = vr;
  dft[NPAD + idx] = vi;
}

// -------- stage A: Xf = F * x * F, one block per (b,i) image --------
__global__ __launch_bounds__(256) void fwd_dft_kernel(const float* __restrict__ x,
                                                      const float* __restrict__ dft,
                                                      float* __restrict__ xr,
                                                      float* __restrict__ xi) {
  extern __shared__ float smem[];
  float* sFr = smem;
  float* sFi = smem + NPAD;
  float* sA  = smem + 2 * NPAD;
  float* sTr = smem + 3 * NPAD;
  float* sTi = smem + 4 * NPAD;

  const float* img = x + (size_t)blockIdx.x * NF;
  for (int t = threadIdx.x; t < NPAD; t += 256) {
    sFr[t] = dft[t];
    sFi[t] = dft[NPAD + t];
    int r = t / LDP, c = t - r * LDP;
    sA[t] = (r < NS && c < NS) ? img[r * NS + c] : 0.f;
  }
  __syncthreads();

  const int wave = threadIdx.x >> 5;
  // T = F * x : 2 components x 16 tiles = 32 jobs over 8 waves
  for (int j = wave; j < 32; j += 8) {
    const int comp = j >> 4, tile = j & 15;
    const int tm = tile >> 2, tn = tile & 3;
    v8f acc = zero8();
    acc = wmma_chain64((comp ? sFi : sFr) + tm * 16 * LDP, LDP, sA + tn * 16, LDP, acc);
    stD_lds((comp ? sTi : sTr) + tm * 16 * LDP + tn * 16, LDP, acc);
  }
  __syncthreads();

  // Xf = T * F : Xr = Tr*Fr - Ti*Fi ; Xi = Tr*Fi + Ti*Fr ; write [f][img] layout
  const int l = threadIdx.x & 31;
  const int n = l & 15, mh = (l >> 4) << 3;
  for (int j = wave; j < 32; j += 8) {
    const int comp = j >> 4, tile = j & 15;
    const int tm = tile >> 2, tn = tile & 3;
    v8f a1 = zero8(), a2 = zero8();
    a1 = wmma_chain64(sTr + tm * 16 * LDP, LDP, (comp ? sFi : sFr) + tn * 16, LDP, a1);
    a2 = wmma_chain64(sTi + tm * 16 * LDP, LDP, (comp ? sFr : sFi) + tn * 16, LDP, a2);
    float* dst = comp ? xi : xr;
#pragma unroll
    for (int v = 0; v < 8; ++v) {
      int row = tm * 16 + mh + v;
      int col = tn * 16 + n;
      float val = comp ? (a1[v] + a2[v]) : (a1[v] - a2[v]);
      if (row < NS && col < NS)
        dst[(size_t)(row * NS + col) * NIMG + blockIdx.x] = val;
    }
  }
}

// -------- stage B: per-frequency complex GEMM [32x64]x[64x128], one block per f --------
__global__ __launch_bounds__(256) void freq_gemm_kernel(const float* __restrict__ xr,
                                                        const float* __restrict__ xi,
                                                        const float* __restrict__ kr,
                                                        const float* __restrict__ ki,
                                                        float* __restrict__ yr,
                                                        float* __restrict__ yi) {
  extern __shared__ float smem[];
  float* sXr = smem;               // [32][68]
  float* sXi = smem + 2176;
  float* sKr = smem + 2 * 2176;    // [64][128]  (B-matrix: [i][o])
  float* sKi = smem + 2 * 2176 + 8192;

  const int f = blockIdx.x;
  for (int t = threadIdx.x; t < NIMG; t += 256) {
    int b = t >> 6, i = t & 63;
    sXr[b * 68 + i] = xr[(size_t)f * NIMG + t];   // coalesced: [f] slice contiguous
    sXi[b * 68 + i] = xi[(size_t)f * NIMG + t];
  }
  for (int t = threadIdx.x; t < 8192; t += 256) { // t = o*64 + i
    int o = t >> 6, i = t & 63;
    sKr[i * 128 + o] = kr[(size_t)t * NF + f];    // L2-resident strided reads
    sKi[i * 128 + o] = ki[(size_t)t * NF + f];
  }
  __syncthreads();

  const int wave = threadIdx.x >> 5;
  const int l = threadIdx.x & 31;
  const int n = l & 15, mh = (l >> 4) << 3;
  // 2 m-tiles x 8 n-tiles x 2 components = 32 jobs
  for (int j = wave; j < 32; j += 8) {
    const int comp = j >> 4, tile = j & 15;
    const int tm = tile >> 3, tn = tile & 7;
    v8f a1 = zero8(), a2 = zero8();
    a1 = wmma_chain64(sXr + tm * 16 * 68, 68, (comp ? sKi : sKr) + tn * 16, 128, a1);
    a2 = wmma_chain64(sXi + tm * 16 * 68, 68, (comp ? sKr : sKi) + tn * 16, 128, a2);
    float* dst = comp ? yi : yr;
#pragma unroll
    for (int v = 0; v < 8; ++v) {
      int brow = tm * 16 + mh + v;  // 0..31
      int ocol = tn * 16 + n;       // 0..127
      float val = comp ? (a1[v] + a2[v]) : (a1[v] - a2[v]);
      dst[((size_t)brow * 128 + ocol) * NF + f] = val;
    }
  }
}

// -------- stage C: y = real(conj(F) * Y * conj(F))/3136 + bias, one block per (b,o) --------
__global__ __launch_bounds__(256) void inv_dft_kernel(const float* __restrict__ yr,
                                                      const float* __restrict__ yi,
                                                      const float* __restrict__ dft,
                                                      const float* __restrict__ bias,
                                                      float* __restrict__ out) {
  extern __shared__ float smem[];
  float* sFr = smem;
  float* sFi = smem + NPAD;
  float* sYr = smem + 2 * NPAD;
  float* sYi = smem + 3 * NPAD;
  float* sUr = smem + 4 * NPAD;
  float* sUi = smem + 5 * NPAD;

  const float* imr = yr + (size_t)blockIdx.x * NF;
  const float* imi = yi + (size_t)blockIdx.x * NF;
  for (int t = threadIdx.x; t < NPAD; t += 256) {
    sFr[t] = dft[t];
    sFi[t] = dft[NPAD + t];
    int r = t / LDP, c = t - r * LDP;
    bool ok = (r < NS && c < NS);
    sYr[t] = ok ? imr[r * NS + c] : 0.f;
    sYi[t] = ok ? imi[r * NS + c] : 0.f;
  }
  __syncthreads();

  const int wave = threadIdx.x >> 5;
  // U = conj(F)*Y : Ur = Fr*Yr + Fi*Yi ; Ui = Fr*Yi - Fi*Yr
  for (int j = wave; j < 32; j += 8) {
    const int comp = j >> 4, tile = j & 15;
    const int tm = tile >> 2, tn = tile & 3;
    v8f a1 = zero8(), a2 = zero8();
    a1 = wmma_chain64(sFr + tm * 16 * LDP, LDP, (comp ? sYi : sYr) + tn * 16, LDP, a1);
    a2 = wmma_chain64(sFi + tm * 16 * LDP, LDP, (comp ? sYr : sYi) + tn * 16, LDP, a2);
    v8f d;
#pragma unroll
    for (int v = 0; v < 8; ++v) d[v] = comp ? (a1[v] - a2[v]) : (a1[v] + a2[v]);
    stD_lds((comp ? sUi : sUr) + tm * 16 * LDP + tn * 16, LDP, d);
  }
  __syncthreads();

  // y = (Ur*Fr + Ui*Fi)/3136 + bias[o]
  const float bval = bias[blockIdx.x & 127];
  const float scale = 1.0f / (float)NF;
  const int l = threadIdx.x & 31;
  const int n = l & 15, mh = (l >> 4) << 3;
  float* dst = out + (size_t)blockIdx.x * NF;
  for (int j = wave; j < 16; j += 8) {
    const int tm = j >> 2, tn = j & 3;
    v8f a1 = zero8(), a2 = zero8();
    a1 = wmma_chain64(sUr + tm * 16 * LDP, LDP, sFr + tn * 16, LDP, a1);
    a2 = wmma_chain64(sUi + tm * 16 * LDP, LDP, sFi + tn * 16, LDP, a2);
#pragma unroll
    for (int v = 0; v < 8; ++v) {
      int row = tm * 16 + mh + v;
      int col = tn * 16 + n;
      if (row < NS && col < NS)
        dst[row * NS + col] = (a1[v] + a2[v]) * scale + bval;
    }
  }
}

extern "C" void kernel_launch(void* const* d_in, const int* in_sizes, int n_in,
                              void* d_out, int out_size, void* d_ws, size_t ws_size,
                              hipStream_t stream) {
  const float* x    = (const float*)d_in[0];  // [32,64,56,56]
  const float* kr   = (const float*)d_in[1];  // [128,64,56,56]
  const float* ki   = (const float*)d_in[2];  // [128,64,56,56]
  const float* bias = (const float*)d_in[3];  // [128]
  float* out = (float*)d_out;                 // [32,128,56,56]

  float* ws  = (float*)d_ws;
  float* dft = ws;                                  // 2*NPAD floats
  float* XR  = ws + 2 * NPAD;                       // NF*NIMG
  float* XI  = XR + (size_t)NF * NIMG;
  float* YR  = XI + (size_t)NF * NIMG;              // NOUT*NF
  float* YI  = YR + (size_t)NOUT * NF;

  const size_t shA = 5 * NPAD * sizeof(float);                 // 87,040 B
  const size_t shB = (2 * 2176 + 2 * 8192) * sizeof(float);    // 82,944 B
  const size_t shC = 6 * NPAD * sizeof(float);                 // 104,448 B
  // gfx1250 allows up to 320KB LDS per workgroup; opt in for >64KB dynamic LDS.
  // Idempotent + deterministic: no static guards (harness contract).
  hipFuncSetAttribute((const void*)fwd_dft_kernel,
                      hipFuncAttributeMaxDynamicSharedMemorySize, (int)shA);
  hipFuncSetAttribute((const void*)freq_gemm_kernel,
                      hipFuncAttributeMaxDynamicSharedMemorySize, (int)shB);
  hipFuncSetAttribute((const void*)inv_dft_kernel,
                      hipFuncAttributeMaxDynamicSharedMemorySize, (int)shC);

  build_dft_kernel<<<(NPAD + 255) / 256, 256, 0, stream>>>(dft);
  fwd_dft_kernel<<<NIMG, 256, shA, stream>>>(x, dft, XR, XI);
  freq_gemm_kernel<<<NF, 256, shB, stream>>>(XR, XI, kr, ki, YR, YI);
  inv_dft_kernel<<<NOUT, 256, shC, stream>>>(YR, YI, dft, bias, out);
}